// HawkesLayer_31782757991226
// MI455X (gfx1250) — compile-verified
//
#include <hip/hip_runtime.h>
#include <math.h>

// Hawkes process, MI455X (gfx1250, wave32).
// Kernel A: bulk copy of the selected training sequence (80 KB) into out[0:20000].
// Kernel B: single block, 10 waves.
//   Phase 1a: one wave per type computes S0[k] (2000-term weighted reduction).
//   Phase 1b: all 320 threads precompute the term1[1000][10] table (pure f(i,k),
//             independent of the recurrence) into LDS — removes all exp/div work
//             from the serial loop.
//   Phase 2 : wave 0 runs the 1000-step serial scan; the 10x10 matvec runs on the
//             matrix pipe via V_WMMA_F32_16X16X4_F32 (alpha padded to 16x16,
//             K split into 3 chunks of 4, C-chained).

typedef __attribute__((ext_vector_type(2))) float v2f;
typedef __attribute__((ext_vector_type(8))) float v8f;

#define NB_EVENT 2000
#define NB_TYPE  10
#define PRED_LEN 1000
#define T1_PAD   16   // inactive lanes may index k up to 15; keep reads in-bounds

__global__ __launch_bounds__(128) void hawkes_copy_train(const int* __restrict__ seq_id,
                                                         const float* __restrict__ sequences,
                                                         float* __restrict__ out) {
    int sid = seq_id[0];
    const float4* __restrict__ src =
        (const float4*)(sequences + (size_t)sid * (NB_EVENT * NB_TYPE));
    float4* __restrict__ dst = (float4*)out;
    const int n4 = NB_EVENT * NB_TYPE / 4;  // 5000
    for (int i = blockIdx.x * blockDim.x + threadIdx.x; i < n4; i += gridDim.x * blockDim.x)
        dst[i] = src[i];
}

__global__ __launch_bounds__(320) void hawkes_scan(const int* __restrict__ seq_id,
                                                   const float* __restrict__ sequences,
                                                   const float* __restrict__ spontaneous,
                                                   const float* __restrict__ theta_g,
                                                   const float* __restrict__ w_g,
                                                   const float* __restrict__ alpha_g,
                                                   float* __restrict__ out) {
    __shared__ float sS0[NB_TYPE];
    __shared__ float sTerm1[PRED_LEN * NB_TYPE + T1_PAD];

    const int tid  = threadIdx.x;
    const int wave = tid >> 5;
    const int lane = tid & 31;
    const int sid  = seq_id[0];
    const float* __restrict__ train = sequences + (size_t)sid * (NB_EVENT * NB_TYPE);

    // ---- Phase 1a: S0[k] = sum_tao train[tao,k] * exp(-w_k * (2000 - tao)); one wave per type
    {
        const int k = wave;  // 0..9 (blockDim = 320 = 10 waves)
        const float wk = w_g[sid * NB_TYPE + k];
        float partial = 0.0f;
        for (int tao = lane; tao < NB_EVENT; tao += 32) {
            float decay = expf(-wk * (float)(NB_EVENT - tao));
            partial = fmaf(train[tao * NB_TYPE + k], decay, partial);
        }
        for (int s = 16; s > 0; s >>= 1) partial += __shfl_down(partial, s, 32);
        if (lane == 0) sS0[k] = partial;
    }

    // ---- Phase 1b: term1 table, term1[i][k] = spont/theta * (e^{-th(2000+i)} - e^{-th(2001+i)})
    for (int idx = tid; idx < PRED_LEN * NB_TYPE; idx += 320) {
        const int i = idx / NB_TYPE;
        const int k = idx - i * NB_TYPE;
        const float th = theta_g[sid * NB_TYPE + k];
        const float so = spontaneous[sid * NB_TYPE + k] / th;
        const float ea = expf(-th * (float)(NB_EVENT + i));
        const float eb = expf(-th * (float)(NB_EVENT + i + 1));
        sTerm1[idx] = so * (ea - eb);
    }
    __syncthreads();
    if (tid >= 32) return;   // wave 0 carries the serial recurrence; EXEC all-ones from here

    // Lane->type mapping chosen so WMMA D rows land in the consuming lane:
    // types 0..7 -> lanes 0..7 (D rows 0..7 live in lanes 0..15),
    // types 8,9  -> lanes 16,17 (D rows 8..15 live in lanes 16..31).
    const int  hi     = (lane >= 16) ? 1 : 0;
    const int  k      = (lane & 7) + 8 * hi;       // inactive lanes: harmless 0..15
    const bool active = ((lane & 15) < 8) && (k < NB_TYPE);

    // Inactive lanes are numerically inert: ew=0 => S stays 0 => effect=0,
    // which also supplies the zero padding for K=10,11 via shuffle sources 18,19.
    float S = 0.0f, ew = 0.0f, omew = 1.0f, invw = 0.0f;
    if (active) {
        S    = sS0[k];
        ew   = expf(-w_g[sid * NB_TYPE + k]);      // exp(-w * DELTA), DELTA = 1
        omew = 1.0f - ew;
        invw = 1.0f / w_g[sid * NB_TYPE + k];
    }

    // A-matrix chunks (16x4 f32, 2 VGPRs each): lane m=lane&15 holds row m;
    // VGPR0 holds K = cb, VGPR1 holds K = cb+1 (cb offset by 2 for the hi half).
    const int m = lane & 15;
    v2f A0 = {0.0f, 0.0f}, A1 = {0.0f, 0.0f}, A2 = {0.0f, 0.0f};
    if (m < NB_TYPE) {
        const float* __restrict__ arow = alpha_g + (size_t)sid * NB_TYPE * NB_TYPE + m * NB_TYPE;
        const int cb0 = 0 + 2 * hi, cb1 = 4 + 2 * hi, cb2 = 8 + 2 * hi;
        A0.x = arow[cb0];  A0.y = arow[cb0 + 1];
        A1.x = arow[cb1];  A1.y = arow[cb1 + 1];
        if (cb2     < NB_TYPE) A2.x = arow[cb2];      // cols 10,11 padded to 0
        if (cb2 + 1 < NB_TYPE) A2.y = arow[cb2 + 1];
    }

    float* __restrict__ pout = out + NB_EVENT * NB_TYPE;
    const int j    = lane & 7;
    const int off2 = 2 * hi;

    for (int i = 0; i < PRED_LEN; ++i) {
        const float effect = S * omew;

        // Broadcast effect chunks into B (4x16, all 16 columns identical).
        // eff index -> source lane: idx<8 -> idx ; idx 8..11 -> lanes 16..19 (18,19 hold 0).
        const float b00 = __shfl(effect, off2 + 0, 32);
        const float b01 = __shfl(effect, off2 + 1, 32);
        const float b10 = __shfl(effect, off2 + 4, 32);
        const float b11 = __shfl(effect, off2 + 5, 32);
        const float b20 = __shfl(effect, off2 + 16, 32);  // eff[8] / eff[10]=0
        const float b21 = __shfl(effect, off2 + 17, 32);  // eff[9] / eff[11]=0
        const v2f B0 = {b00, b01}, B1 = {b10, b11}, B2 = {b20, b21};

        // y(16) = alpha_pad(16x16) @ effect(16), K chunked 3x4, f32-exact.
        v8f acc = {0.f, 0.f, 0.f, 0.f, 0.f, 0.f, 0.f, 0.f};
        acc = __builtin_amdgcn_wmma_f32_16x16x4_f32(false, A0, false, B0, (short)0, acc, false, false);
        acc = __builtin_amdgcn_wmma_f32_16x16x4_f32(false, A1, false, B1, (short)0, acc, false, false);
        acc = __builtin_amdgcn_wmma_f32_16x16x4_f32(false, A2, false, B2, (short)0, acc, false, false);

        // Extract y[k]: all D columns identical, so this lane's half holds the
        // needed row in acc[j]; select with a cndmask tree (no LDS/shuffles).
        const float y01 = (j & 1) ? acc[1] : acc[0];
        const float y23 = (j & 1) ? acc[3] : acc[2];
        const float y45 = (j & 1) ? acc[5] : acc[4];
        const float y67 = (j & 1) ? acc[7] : acc[6];
        const float ya  = (j & 2) ? y23 : y01;
        const float yb  = (j & 2) ? y67 : y45;
        const float y   = (j & 4) ? yb : ya;

        const float term1 = sTerm1[i * NB_TYPE + k];   // precomputed; off the dep chain
        const float pred  = fmaf(y, invw, term1);
        if (active) pout[i * NB_TYPE + k] = pred;
        S = ew * (S + pred);
    }
}

extern "C" void kernel_launch(void* const* d_in, const int* in_sizes, int n_in,
                              void* d_out, int out_size, void* d_ws, size_t ws_size,
                              hipStream_t stream) {
    (void)in_sizes; (void)n_in; (void)out_size; (void)d_ws; (void)ws_size;
    const int*   seq_id      = (const int*)d_in[0];
    const float* sequences   = (const float*)d_in[1];
    const float* spontaneous = (const float*)d_in[2];
    const float* theta       = (const float*)d_in[3];
    const float* w           = (const float*)d_in[4];
    const float* alpha       = (const float*)d_in[5];
    float*       out         = (float*)d_out;

    hawkes_copy_train<<<40, 128, 0, stream>>>(seq_id, sequences, out);
    hawkes_scan<<<1, 320, 0, stream>>>(seq_id, sequences, spontaneous, theta, w, alpha, out);
}